// ContrastiveLoss_58901181497764
// MI455X (gfx1250) — compile-verified
//
#include <hip/hip_runtime.h>

typedef __attribute__((ext_vector_type(2))) float v2f;
typedef __attribute__((ext_vector_type(8))) float v8f;

// Types for the async global->LDS DMA builtin (param types per hipcc diagnostic:
// 'int __attribute__((vector_size(16))) __device__ *' i.e. v4i in addrspace(1)).
typedef int v4i_t __attribute__((vector_size(16)));
typedef v4i_t __attribute__((address_space(1)))* g_v4i_p;   // global
typedef v4i_t __attribute__((address_space(3)))* l_v4i_p;   // LDS

#define NB 16384
#define DD 256
#define NCHUNK 4
#define TILES_PER_CHUNK 256   // 1024 column tiles / 4 chunks
#define LDS_STRIDE 260        // 256 + 4 pad floats -> conflict-free ds_load_b64

#if defined(__has_builtin)
#if __has_builtin(__builtin_amdgcn_global_load_async_to_lds_b128)
#define HAVE_ASYNC_LDS 1
#endif
#endif

// ---------------------------------------------------------------------------
// Zero the scalar output (d_out is poisoned by the harness).
__global__ void cl_zero(float* out) { out[0] = 0.f; }

// ---------------------------------------------------------------------------
// Row norms: one wave per row, 2 x float4 per lane, shfl reduction.
__global__ __launch_bounds__(256) void cl_norms(const float* __restrict__ O,
                                                const float* __restrict__ T,
                                                float* __restrict__ oinv,
                                                float* __restrict__ tinv) {
  int gw   = (blockIdx.x * 256 + threadIdx.x) >> 5;   // global wave id
  int lane = threadIdx.x & 31;
  const float* src; float* dst; int r;
  if (gw < NB) { src = O; dst = oinv; r = gw; }
  else         { src = T; dst = tinv; r = gw - NB; }
  const float4* p = reinterpret_cast<const float4*>(src + (size_t)r * DD);
  float ss = 0.f;
#pragma unroll
  for (int i = 0; i < 2; ++i) {
    float4 v = p[lane + 32 * i];
    ss += v.x * v.x + v.y * v.y + v.z * v.z + v.w * v.w;
  }
#pragma unroll
  for (int off = 16; off > 0; off >>= 1) ss += __shfl_xor(ss, off, 32);
  if (lane == 0) dst[r] = rsqrtf(ss);
}

// ---------------------------------------------------------------------------
// Fused cosine-logits + online logsumexp.
//   grid = (NB/128, NCHUNK), block = 256 (8 waves).
//   Each wave: 16 rows of O (A tile held in 128 VGPRs).
//   Block stages one 16-row T tile (B) in LDS per iteration, double buffered
//   via GLOBAL_LOAD_ASYNC_TO_LDS_B128 (no VGPR staging, ASYNCcnt-tracked).
//   Per tile per wave: 64 x V_WMMA_F32_16X16X4_F32 over K=256.
__global__ __launch_bounds__(256) void cl_main(const float* __restrict__ O,
                                               const float* __restrict__ T,
                                               const float* __restrict__ oinv,
                                               const float* __restrict__ tinv,
                                               float* __restrict__ mpart,
                                               float* __restrict__ spart,
                                               float* __restrict__ diagv) {
  __shared__ float lb[2][16 * LDS_STRIDE];

  const int tid   = threadIdx.x;
  const int lane  = tid & 31;
  const int wv    = tid >> 5;
  const int l16   = lane & 15;
  const int hi    = lane >> 4;          // 0 or 1 (half-wave)
  const int chunk = blockIdx.y;
  const int j0    = chunk * TILES_PER_CHUNK;
  const int rowbase = blockIdx.x * 128 + wv * 16;
  const int jdiag   = rowbase >> 4;     // column tile containing the diagonal

  // ---- A tile (16 rows x 256 K) in registers: lane L holds
  //      O[rowbase + L%16][4k + 2*(L/16) .. +1] for k = 0..63 ----
  v2f a[64];
  {
    const float2* ap = reinterpret_cast<const float2*>(
        O + (size_t)(rowbase + l16) * DD + 2 * hi);
#pragma unroll
    for (int ks = 0; ks < 64; ++ks) {
      float2 t = ap[2 * ks];
      a[ks].x = t.x; a[ks].y = t.y;
    }
  }

  float osc[8];
#pragma unroll
  for (int r = 0; r < 8; ++r) osc[r] = oinv[rowbase + r + 8 * hi];

  float m[8], s[8];
#pragma unroll
  for (int r = 0; r < 8; ++r) { m[r] = -3.0e38f; s[r] = 0.f; }

  // ---- cooperative B-tile staging: 16 rows x 256 f32 = 1024 x b128 ----
  // Async path: DMA straight into LDS, no VGPR staging, no spill.
  auto prefetch = [&](int jg, int buf) {
    const float* src = T + (size_t)jg * 16 * DD;
#pragma unroll
    for (int p = 0; p < 4; ++p) {
      int idx = tid + 256 * p;
      int rr  = idx >> 6;
      int cc  = (idx & 63) << 2;
#if defined(HAVE_ASYNC_LDS)
      __builtin_amdgcn_global_load_async_to_lds_b128(
          (g_v4i_p)(src + rr * DD + cc),
          (l_v4i_p)&lb[buf][rr * LDS_STRIDE + cc],
          0, 0);
#else
      float4 v = *reinterpret_cast<const float4*>(src + rr * DD + cc);
      *reinterpret_cast<float4*>(&lb[buf][rr * LDS_STRIDE + cc]) = v;
#endif
    }
  };
  auto wait_stage = [&]() {
#if defined(HAVE_ASYNC_LDS)
    __builtin_amdgcn_s_wait_asynccnt(0);
#endif
    __syncthreads();
  };

  prefetch(j0, 0);
  wait_stage();

  for (int j = 0; j < TILES_PER_CHUNK; ++j) {
    const int buf = j & 1;
    const int jg  = j0 + j;
    if (j + 1 < TILES_PER_CHUNK) prefetch(jg + 1, buf ^ 1);  // in flight during WMMA

    v8f acc = {0.f, 0.f, 0.f, 0.f, 0.f, 0.f, 0.f, 0.f};
    const float2* bp = reinterpret_cast<const float2*>(
        &lb[buf][l16 * LDS_STRIDE + 2 * hi]);
#pragma unroll
    for (int ks = 0; ks < 64; ++ks) {
      float2 t = bp[2 * ks];                      // ds_load b64, bank-free
      v2f b; b.x = t.x; b.y = t.y;
      acc = __builtin_amdgcn_wmma_f32_16x16x4_f32(
          false, a[ks], false, b, (short)0, acc, false, false);
    }

    // scale to cosine logits + online logsumexp; D layout:
    //   acc[r] = logits[rowbase + r + 8*hi][jg*16 + l16]
    const float tsc = tinv[jg * 16 + l16];
#pragma unroll
    for (int r = 0; r < 8; ++r) {
      float logit = acc[r] * osc[r] * tsc;
      if (jg == jdiag && l16 == r + 8 * hi) diagv[rowbase + l16] = logit;
      float mn = fmaxf(m[r], logit);
      s[r] = s[r] * __expf(m[r] - mn) + __expf(logit - mn);
      m[r] = mn;
    }

    wait_stage();   // async loads for j+1 done; all waves done reading buf
  }

  // ---- reduce (m,s) across the 16 lanes of each half-wave (cols) ----
#pragma unroll
  for (int r = 0; r < 8; ++r) {
#pragma unroll
    for (int off = 1; off < 16; off <<= 1) {      // stays within 16-lane half
      float mo = __shfl_xor(m[r], off, 32);
      float so = __shfl_xor(s[r], off, 32);
      float mn = fmaxf(m[r], mo);
      s[r] = s[r] * __expf(m[r] - mn) + so * __expf(mo - mn);
      m[r] = mn;
    }
  }
  if (l16 == 0) {
#pragma unroll
    for (int r = 0; r < 8; ++r) {
      int row = rowbase + 8 * hi + r;
      mpart[(size_t)chunk * NB + row] = m[r];
      spart[(size_t)chunk * NB + row] = s[r];
    }
  }
}

// ---------------------------------------------------------------------------
// Merge the NCHUNK partial logsumexps per row, subtract diag, mean-reduce.
__global__ __launch_bounds__(256) void cl_combine(const float* __restrict__ mpart,
                                                  const float* __restrict__ spart,
                                                  const float* __restrict__ diagv,
                                                  float* __restrict__ out) {
  int row = blockIdx.x * 256 + threadIdx.x;
  float m = -3.0e38f, s = 0.f;
#pragma unroll
  for (int c = 0; c < NCHUNK; ++c) {
    float mc = mpart[(size_t)c * NB + row];
    float sc = spart[(size_t)c * NB + row];
    float mn = fmaxf(m, mc);
    s = s * __expf(m - mn) + sc * __expf(mc - mn);
    m = mn;
  }
  float v = m + __logf(s) - diagv[row];
#pragma unroll
  for (int off = 16; off > 0; off >>= 1) v += __shfl_xor(v, off, 32);
  if ((threadIdx.x & 31) == 0) atomicAdd(out, v * (1.0f / NB));
}

// ---------------------------------------------------------------------------
extern "C" void kernel_launch(void* const* d_in, const int* in_sizes, int n_in,
                              void* d_out, int out_size, void* d_ws, size_t ws_size,
                              hipStream_t stream) {
  const float* O = (const float*)d_in[0];   // outputs [16384, 256] f32
  const float* T = (const float*)d_in[1];   // targets [16384, 256] f32
  float* out = (float*)d_out;               // scalar f32 loss
  float* ws  = (float*)d_ws;
  float* oinv  = ws;                         // [NB]
  float* tinv  = ws + NB;                    // [NB]
  float* mpart = ws + 2 * NB;                // [NCHUNK, NB]
  float* spart = ws + (2 + NCHUNK) * NB;     // [NCHUNK, NB]
  float* diagv = ws + (2 + 2 * NCHUNK) * NB; // [NB]   (total ~704 KB)

  cl_zero<<<1, 1, 0, stream>>>(out);
  cl_norms<<<(2 * NB) / 8, 256, 0, stream>>>(O, T, oinv, tinv);
  cl_main<<<dim3(NB / 128, NCHUNK), 256, 0, stream>>>(O, T, oinv, tinv,
                                                      mpart, spart, diagv);
  cl_combine<<<NB / 256, 256, 0, stream>>>(mpart, spart, diagv, out);
}